// E3nnForce_15960098472106
// MI455X (gfx1250) — compile-verified
//
#include <hip/hip_runtime.h>
#include <math.h>

typedef float v2f __attribute__((ext_vector_type(2)));
typedef float v8f __attribute__((ext_vector_type(8)));

#define C_      50
#define HIDDEN_ 64
#define LAYERS_ 6
#define NB_     10
#define TT_     512
#define LL_     514
#define NG      2            // groups per block
#define NE      (NG*8)       // 16 edges
#define NN      (NG*9)       // 18 nodes
#define LDC     64           // padded channel dim (unguarded stores)
#define ROWS    32           // padded node rows (2 M-tiles)

__device__ __forceinline__ float sigmoidf_(float x) { return 1.0f / (1.0f + __expf(-x)); }
__device__ __forceinline__ float siluf_(float x)    { return x * sigmoidf_(x); }

__device__ __forceinline__ v8f wmma4(v2f a, v2f b, v8f c) {
  // V_WMMA_F32_16X16X4_F32 : D = A(16x4,f32) * B(4x16,f32) + C(16x16,f32)
  return __builtin_amdgcn_wmma_f32_16x16x4_f32(false, a, false, b, (short)0, c, false, false);
}

__global__ __launch_bounds__(256)
void e3nn_fused(const float* __restrict__ y,
                const float* __restrict__ Wemb,
                const float* __restrict__ Wr1,
                const float* __restrict__ br1,
                const float* __restrict__ Wr2,
                const float* __restrict__ Ws,
                const float* __restrict__ Wv,
                const float* __restrict__ Wg,
                const float* __restrict__ Wout,
                float* __restrict__ res)
{
  __shared__ float sS[ROWS][LDC];          // scalar features (M-pad rows stay 0)
  __shared__ float sV[3][ROWS][LDC];       // vector features, per component
  __shared__ float sH[ROWS][LDC];          // h = s + s_agg
  __shared__ float sHid[NE][HIDDEN_];      // edge MLP hidden
  __shared__ float sW[NE][256];            // edge path weights; also Wr1/br1 staging
  __shared__ float sU[NE][4];              // ux, uy, uz, fcut
  __shared__ float sEmb[NE][12];           // radial basis * fcut + edge attrs
  __shared__ float sY2[NE][9];             // u u^T - I/3
  __shared__ float sPos[NN][2];
  __shared__ float sOut[NG][3][2];

  const int tid  = threadIdx.x;
  const int lane = tid & 31;
  const int wv   = tid >> 5;     // wave id 0..7
  const int half = lane >> 4;    // lane half selects K pair / M half
  const int l15  = lane & 15;
  const int g0   = blockIdx.x * NG;

  // message-phase decomposition: 16 threads per edge, shift/mask only
  const int eM   = tid >> 4;     // edge 0..15
  const int cB   = tid & 15;     // channel base lane
  const int leM  = eM & 7;
  const int srcM = 9 * (eM >> 3) + (int)((0x41774411u >> (leM * 4)) & 0xF);
  const int dstM = 9 * (eM >> 3) + (int)((0x74865320u >> (leM * 4)) & 0xF);

  // zero state planes once (M-pad rows must start zero and self-maintain)
  for (int i = tid; i < ROWS * LDC; i += 256) {
    (&sS[0][0])[i] = 0.f; (&sH[0][0])[i] = 0.f;
    (&sV[0][0][0])[i] = 0.f; (&sV[0][0][0])[i + ROWS*LDC] = 0.f; (&sV[0][0][0])[i + 2*ROWS*LDC] = 0.f;
  }

  // --- geometry: 9 points per group (3 bars x {+off, ctr, -off}) ---
  if (tid < NN) {
    int i = tid / 9, p = tid % 9, bar = p / 3, j = p % 3;
    int g = g0 + i, b = g / TT_, t = g % TT_;
    const float* yy = y + ((size_t)b * LL_ + (t + bar)) * 6;
    float cx = yy[0], cy = yy[1], a = yy[2];
    float sa, ca; __sincosf(a, &sa, &ca);
    float f = (float)(1 - j);                 // +1, 0, -1
    sPos[tid][0] = cx + f * (-0.05f * sa);
    sPos[tid][1] = cy + f * ( 0.05f * ca);
  }
  __syncthreads();

  // --- per-edge: u, fcut, radial basis embedding, Y2 ---
  if (tid < NE) {
    int i = tid >> 3, le = tid & 7;
    int srcl = 9 * i + (int)((0x41774411u >> (le * 4)) & 0xF);
    int dstl = 9 * i + (int)((0x74865320u >> (le * 4)) & 0xF);
    float ex = sPos[srcl][0] - sPos[dstl][0];
    float ey = sPos[srcl][1] - sPos[dstl][1];
    float r  = sqrtf(ex * ex + ey * ey);
    float inv = 1.0f / (r + 1e-12f);
    float ux = ex * inv, uy = ey * inv;
    float tcl = fminf(fmaxf(r / 0.06f, 0.f), 1.f);
    float fcut = 0.5f * (__cosf(3.14159265358979f * tcl) + 1.f);
    sU[tid][0] = ux; sU[tid][1] = uy; sU[tid][2] = 0.f; sU[tid][3] = fcut;
    const float width = 0.06f / 9.0f;
    for (int k = 0; k < NB_; ++k) {
      float ck = 0.06f * (float)k / 9.0f;
      float zz = (r - ck) / width;
      sEmb[tid][k] = __expf(-zz * zz) * fcut;
    }
    sEmb[tid][10] = (le < 6) ? 0.f : 1.f;   // edge_attr one-hot
    sEmb[tid][11] = (le < 6) ? 1.f : 0.f;
    sY2[tid][0] = ux*ux - (1.f/3.f); sY2[tid][1] = ux*uy;           sY2[tid][2] = 0.f;
    sY2[tid][3] = ux*uy;             sY2[tid][4] = uy*uy-(1.f/3.f); sY2[tid][5] = 0.f;
    sY2[tid][6] = 0.f;               sY2[tid][7] = 0.f;             sY2[tid][8] = -(1.f/3.f);
  }

  // --- s init: [1, node_attr] @ W_embed ---
  for (int idx = tid; idx < NN * C_; idx += 256) {
    int n = idx / C_, c = idx % C_;
    int j = (n % 9) % 3;
    int row = (j == 1) ? 1 : 2;
    sS[n][c] = Wemb[c] + Wemb[row * C_ + c];
  }
  __syncthreads();

  for (int l = 0; l < LAYERS_; ++l) {
    const float* Wr1l = Wr1 + l * 12 * HIDDEN_;
    const float* br1l = br1 + l * HIDDEN_;
    const float* Wr2l = Wr2 + l * HIDDEN_ * 250;
    const float* Wsl  = Ws  + l * C_ * C_;
    const float* Wvl  = Wv  + l * C_ * C_;
    const float* Wgl  = Wg  + l * C_ * C_;

    // stage Wr1 (+br1) into the sW region (free: previous layer's w consumed)
    float* wr1s = &sW[0][0];            // 768 floats + 64 bias
    for (int i = tid; i < 12 * HIDDEN_; i += 256) wr1s[i] = Wr1l[i];
    if (tid < HIDDEN_) wr1s[12 * HIDDEN_ + tid] = br1l[tid];
    __syncthreads();

    // edge MLP layer 1 (K=12, tiny -> VALU): hidden = silu(emb @ Wr1 + br1)
    for (int idx = tid; idx < NE * HIDDEN_; idx += 256) {
      int e = idx >> 6, hh = idx & 63;
      float acc = wr1s[12 * HIDDEN_ + hh];
      #pragma unroll
      for (int k = 0; k < 12; ++k) acc += sEmb[e][k] * wr1s[k * HIDDEN_ + hh];
      sHid[e][hh] = siluf_(acc);
    }
    __syncthreads();

    // edge GEMM via fp32 WMMA: hidden(16x64) @ Wr2(64x250->pad256).
    // Column clamp hoisted out of the loop; garbage lands only in sW columns
    // 250..255, which the message phase never reads.
    {
      v8f acc0 = {0,0,0,0,0,0,0,0};
      v8f acc1 = {0,0,0,0,0,0,0,0};
      const int na  = wv * 16 + l15;                  // always < 250
      const int nb  = (wv + 8) * 16 + l15;            // 128..255
      const int nbc = min(nb, 249);                   // hoisted clamp
      for (int ks = 0; ks < 16; ++ks) {
        int kk = ks * 4 + 2 * half;
        v2f a; a.x = sHid[l15][kk]; a.y = sHid[l15][kk + 1];
        v2f b0, b1;
        b0.x = Wr2l[kk * 250 + na];  b0.y = Wr2l[(kk + 1) * 250 + na];
        b1.x = Wr2l[kk * 250 + nbc]; b1.y = Wr2l[(kk + 1) * 250 + nbc];
        acc0 = wmma4(a, b0, acc0);
        acc1 = wmma4(a, b1, acc1);
      }
      #pragma unroll
      for (int r = 0; r < 8; ++r) {
        int m = r + 8 * half;               // edge row 0..15
        float fc = sU[m][3];                // w *= fcut
        sW[m][na] = acc0[r] * fc;
        sW[m][nb] = acc1[r] * fc;
      }
    }
    // h <- s: flat float4 copy of the whole [32][64] plane (pad cols of sH may
    // be garbage: they are only read in the masked epilogue k-step)
    {
      const float4* src4 = (const float4*)&sS[0][0];
      float4*       dst4 = (float4*)&sH[0][0];
      dst4[tid]       = src4[tid];
      dst4[tid + 256] = src4[tid + 256];
    }
    __syncthreads();

    // messages: 16 threads per edge, c = cB + 16*it (shift/mask only, no div).
    // Read phase into registers; src reads must complete before dst writes.
    float mreg[4][4];
    {
      float ux = sU[eM][0], uy = sU[eM][1], uz = sU[eM][2];
      float q0 = sY2[eM][0], q1 = sY2[eM][1], q2 = sY2[eM][2];
      float q3 = sY2[eM][3], q4 = sY2[eM][4], q5 = sY2[eM][5];
      float q6 = sY2[eM][6], q7 = sY2[eM][7], q8 = sY2[eM][8];
      #pragma unroll
      for (int it = 0; it < 4; ++it) {
        int c  = cB + 16 * it;
        int cc = min(c, C_ - 1);            // clamped read index (writes guarded)
        float w0 = sW[eM][0*C_ + cc], w1 = sW[eM][1*C_ + cc], w2 = sW[eM][2*C_ + cc];
        float w3 = sW[eM][3*C_ + cc], w4 = sW[eM][4*C_ + cc];
        float ssv = sS[srcM][cc];
        float v0 = sV[0][srcM][cc], v1 = sV[1][srcM][cc], v2 = sV[2][srcM][cc];
        float dotvu = v0 * ux + v1 * uy + v2 * uz;
        float cr0 = v1 * uz - v2 * uy;
        float cr1 = v2 * ux - v0 * uz;
        float cr2 = v0 * uy - v1 * ux;
        float y0  = v0 * q0 + v1 * q3 + v2 * q6;
        float y1  = v0 * q1 + v1 * q4 + v2 * q7;
        float y2c = v0 * q2 + v1 * q5 + v2 * q8;
        mreg[it][0] = w0 * ssv + w2 * dotvu;
        mreg[it][1] = w1 * ssv * ux + w3 * cr0 + w4 * y0;
        mreg[it][2] = w1 * ssv * uy + w3 * cr1 + w4 * y1;
        mreg[it][3] = w1 * ssv * uz + w3 * cr2 + w4 * y2c;
      }
    }
    __syncthreads();
    // write phase: each dst node receives exactly one edge -> race free
    #pragma unroll
    for (int it = 0; it < 4; ++it) {
      int c = cB + 16 * it;
      if (c < C_) {
        sH[dstM][c]    += mreg[it][0];
        sV[0][dstM][c] += mreg[it][1];
        sV[1][dstM][c] += mreg[it][2];
        sV[2][dstM][c] += mreg[it][3];
      }
    }
    __syncthreads();

    // node GEMMs, all 5 fused in one k-loop (shared A for Ws/Wg, shared B for
    // the 3 Wv planes). Main 12 steps are clamp/mask free: kk<=47 keeps every
    // index in bounds; garbage only reaches D pad columns (allowed). Epilogue
    // handles K=48,49 with a hoisted column clamp and lane-half mask, which
    // also neutralizes the garbage A columns 50,51.
    {
      const int mt = wv & 1, ntile = wv >> 1;   // 2 M-tiles x 4 N-tiles = 8 waves
      const int n = ntile * 16 + l15;
      const int arow = mt * 16 + l15;
      v8f accS  = {0,0,0,0,0,0,0,0};
      v8f accG  = {0,0,0,0,0,0,0,0};
      v8f accV0 = {0,0,0,0,0,0,0,0};
      v8f accV1 = {0,0,0,0,0,0,0,0};
      v8f accV2 = {0,0,0,0,0,0,0,0};
      for (int ks = 0; ks < 12; ++ks) {
        int kk = ks * 4 + 2 * half;
        v2f aH; aH.x = sH[arow][kk];    aH.y = sH[arow][kk + 1];
        v2f a0; a0.x = sV[0][arow][kk]; a0.y = sV[0][arow][kk + 1];
        v2f a1; a1.x = sV[1][arow][kk]; a1.y = sV[1][arow][kk + 1];
        v2f a2; a2.x = sV[2][arow][kk]; a2.y = sV[2][arow][kk + 1];
        int i0 = kk * C_ + n, i1 = i0 + C_;     // <= 2413, in bounds
        v2f bs; bs.x = Wsl[i0]; bs.y = Wsl[i1];
        v2f bg; bg.x = Wgl[i0]; bg.y = Wgl[i1];
        v2f bv; bv.x = Wvl[i0]; bv.y = Wvl[i1];
        accS  = wmma4(aH, bs, accS);
        accG  = wmma4(aH, bg, accG);
        accV0 = wmma4(a0, bv, accV0);
        accV1 = wmma4(a1, bv, accV1);
        accV2 = wmma4(a2, bv, accV2);
      }
      { // epilogue: K = 48,49 live on half=0 lanes; half=1 lanes masked to zero
        const float km = (half == 0) ? 1.f : 0.f;
        const int nc = min(n, C_ - 1);
        int kk = 48 + 2 * half;                 // 48 or 50 (cols 50,51 are dead)
        v2f aH; aH.x = sH[arow][kk];    aH.y = sH[arow][kk + 1];
        v2f a0; a0.x = sV[0][arow][kk]; a0.y = sV[0][arow][kk + 1];
        v2f a1; a1.x = sV[1][arow][kk]; a1.y = sV[1][arow][kk + 1];
        v2f a2; a2.x = sV[2][arow][kk]; a2.y = sV[2][arow][kk + 1];
        v2f bs; bs.x = Wsl[48 * C_ + nc] * km; bs.y = Wsl[49 * C_ + nc] * km;
        v2f bg; bg.x = Wgl[48 * C_ + nc] * km; bg.y = Wgl[49 * C_ + nc] * km;
        v2f bv; bv.x = Wvl[48 * C_ + nc] * km; bv.y = Wvl[49 * C_ + nc] * km;
        accS  = wmma4(aH, bs, accS);
        accG  = wmma4(aH, bg, accG);
        accV0 = wmma4(a0, bv, accV0);
        accV1 = wmma4(a1, bv, accV1);
        accV2 = wmma4(a2, bv, accV2);
      }
      __syncthreads();   // all reads of sH/sV done before any writes
      #pragma unroll
      for (int r = 0; r < 8; ++r) {
        int m = mt * 16 + r + 8 * half;
        float gate = sigmoidf_(accG[r]);
        sS[m][n]    = siluf_(accS[r]);
        sV[0][m][n] = accV0[r] * gate;
        sV[1][m][n] = accV1[r] * gate;
        sV[2][m][n] = accV2[r] * gate;
      }
    }
    __syncthreads();
  }

  // --- readout: out[n,d] = sum_c v[n,c,d]*W_out[c], only bar=1 nodes, d in {0,1} ---
  if (tid < NG * 3 * 2) {
    int i = tid / 6, r2 = tid % 6, jj = r2 >> 1, d = r2 & 1;
    int p = 9 * i + 3 + jj;                  // bar=1 nodes: 3,4,5
    float acc = 0.f;
    for (int c = 0; c < C_; ++c) acc += sV[d][p][c] * Wout[c];
    sOut[i][jj][d] = acc;
  }
  __syncthreads();
  if (tid < NG) {
    int g = g0 + tid, b = g / TT_, t = g % TT_;
    float fx = sOut[tid][0][0] + sOut[tid][1][0] + sOut[tid][2][0];
    float fy = sOut[tid][0][1] + sOut[tid][1][1] + sOut[tid][2][1];
    int p1 = 9 * tid + 4;
    float a0x = sPos[9 * tid + 3][0] - sPos[p1][0];
    float a0y = sPos[9 * tid + 3][1] - sPos[p1][1];
    float a2x = sPos[9 * tid + 5][0] - sPos[p1][0];
    float a2y = sPos[9 * tid + 5][1] - sPos[p1][1];
    float tq = a0x * sOut[tid][0][1] - a0y * sOut[tid][0][0]
             + a2x * sOut[tid][2][1] - a2y * sOut[tid][2][0];
    float* op = res + ((size_t)b * LL_ + (t + 1)) * 3;
    op[0] = fx; op[1] = fy; op[2] = tq;
    if (t == 0)        { float* z = res + (size_t)b * LL_ * 3;              z[0]=0.f; z[1]=0.f; z[2]=0.f; }
    if (t == TT_ - 1)  { float* z = res + ((size_t)b * LL_ + LL_ - 1) * 3;  z[0]=0.f; z[1]=0.f; z[2]=0.f; }
  }
}

extern "C" void kernel_launch(void* const* d_in, const int* in_sizes, int n_in,
                              void* d_out, int out_size, void* d_ws, size_t ws_size,
                              hipStream_t stream) {
  (void)in_sizes; (void)n_in; (void)d_ws; (void)ws_size; (void)out_size;
  const float* y     = (const float*)d_in[0];
  const float* Wemb  = (const float*)d_in[1];
  const float* Wr1   = (const float*)d_in[2];
  const float* br1   = (const float*)d_in[3];
  const float* Wr2   = (const float*)d_in[4];
  const float* Ws    = (const float*)d_in[5];
  const float* Wv    = (const float*)d_in[6];
  const float* Wg    = (const float*)d_in[7];
  const float* Wout  = (const float*)d_in[8];
  float* res = (float*)d_out;

  const int nt = 64 * TT_;                 // 32768 groups
  const int blocks = nt / NG;              // 16384 blocks, 2 groups each
  e3nn_fused<<<blocks, 256, 0, stream>>>(y, Wemb, Wr1, br1, Wr2, Ws, Wv, Wg, Wout, res);
}